// CIN_13958643712071
// MI455X (gfx1250) — compile-verified
//
#include <hip/hip_runtime.h>
#include <hip/hip_bf16.h>

typedef __attribute__((ext_vector_type(2))) float v2f;
typedef __attribute__((ext_vector_type(8))) float v8f;

// One fused layer:  pre[b,o,d] = sum_{h,f} W[o,h*64+f] * X[b,h,d] * In[b,f,d]
//                   x = relu(pre + bias);  Out[b, outOff+o] = sum_d x;  Xout = x
// Block: 256 threads = 8 waves. Each wave owns one 16-row M-tile of the 128
// outputs and 4 accumulators = {b0,b1} x {d 0..15, d 16..31}. Block handles 2
// batches completely (full D=32), so the d-reduction is intra-wave (no atomics).
template <int H, bool STORE_X>
__global__ __launch_bounds__(256) void fused_bilinear_layer(
    const float* __restrict__ Xprev,  // [B, H, 32]
    const float* __restrict__ In,     // [B, 64, 32]
    const float* __restrict__ W,      // [128, H*64] row-major
    const float* __restrict__ bias,   // [128]
    float* __restrict__ Xout,         // [B, 128, 32] (ignored if !STORE_X)
    float* __restrict__ Out,          // [B, 384]
    int outOff) {
  constexpr int K = H * 64;
  constexpr int OUTC = 128;
  constexpr int D = 32;

  __shared__ float xs[2 * H * D];    // x slices for 2 batches
  __shared__ float ins[2 * 64 * D];  // input slices for 2 batches

  const int tid = threadIdx.x;
  const int b0 = blockIdx.x * 2;  // this block's first batch

  // --- cooperative LDS fill (both batches are contiguous in memory) ---
  {
    const float4* src = (const float4*)(Xprev + (size_t)b0 * H * D);
    float4* dst = (float4*)xs;
#pragma unroll 2
    for (int i = tid; i < (2 * H * D) / 4; i += 256) dst[i] = src[i];
  }
  {
    const float4* src = (const float4*)(In + (size_t)b0 * 64 * D);
    float4* dst = (float4*)ins;
#pragma unroll 2
    for (int i = tid; i < (2 * 64 * D) / 4; i += 256) dst[i] = src[i];
  }
  __syncthreads();

  const int lane = tid & 31;
  const int wave = tid >> 5;
  const int mbase = wave * 16;     // M-tile base (output channel)
  const int m = lane & 15;         // A: row within tile / B,C: column (d)
  const int khalf = lane >> 4;     // A/B fragment K-half selector

  v8f acc[4] = {};  // t = bi*2 + dhalf

  // Each lane streams one contiguous W row: W[mbase+m, k + khalf*2 ...]
  const float* wrow = W + (size_t)(mbase + m) * K + khalf * 2;

  for (int h = 0; h < H; ++h) {
    // x value is constant across the whole f-sweep for this h
    float xv[4];
#pragma unroll
    for (int t = 0; t < 4; ++t) {
      const int bi = t >> 1;
      const int d = m + (t & 1) * 16;
      xv[t] = xs[bi * H * D + h * D + d];
    }
#pragma unroll
    for (int f0 = 0; f0 < 64; f0 += 4) {
      const int k = h * 64 + f0;
      // A fragment: 16x4 tile of W (lanes 0-15: K=0,1 ; lanes 16-31: K=2,3)
      const v2f a = *(const v2f*)(wrow + k);
      const int f = f0 + khalf * 2;
#pragma unroll
      for (int t = 0; t < 4; ++t) {
        const int bi = t >> 1;
        const int d = m + (t & 1) * 16;
        // B fragment: G[(h,f),d] = x[h,d] * in[f,d], built on the fly
        const float i0 = ins[bi * 64 * D + f * D + d];
        const float i1 = ins[bi * 64 * D + (f + 1) * D + d];
        v2f b;
        b.x = xv[t] * i0;
        b.y = xv[t] * i1;
        acc[t] = __builtin_amdgcn_wmma_f32_16x16x4_f32(
            false, a, false, b, (short)0, acc[t], false, false);
      }
    }
  }

  // --- epilogue: bias + relu, store x, reduce over d, store out ---
  // C layout: VGPR r, lanes 0-15 -> M = r, lanes 16-31 -> M = r+8; N = lane%16
#pragma unroll
  for (int r = 0; r < 8; ++r) {
    const int o = mbase + r + khalf * 8;
    const float bv = bias[o];
    float s0 = 0.f, s1 = 0.f;  // per-batch partial sums over this lane's 2 d's
#pragma unroll
    for (int t = 0; t < 4; ++t) {
      float v = acc[t][r] + bv;
      v = v > 0.f ? v : 0.f;
      const int bi = t >> 1;
      const int d = m + (t & 1) * 16;
      if (STORE_X) {
        Xout[(size_t)(b0 + bi) * OUTC * D + (size_t)o * D + d] = v;
      }
      if (bi == 0) s0 += v; else s1 += v;
    }
    // butterfly reduction across the 16 lanes that share this o
#pragma unroll
    for (int sh = 1; sh < 16; sh <<= 1) {
      s0 += __shfl_xor(s0, sh, 32);
      s1 += __shfl_xor(s1, sh, 32);
    }
    if (m == 0) {  // lanes 0 and 16 hold the two o's full sums
      Out[(size_t)b0 * 384 + outOff + o] = s0;
      Out[(size_t)(b0 + 1) * 384 + outOff + o] = s1;
    }
  }
}

extern "C" void kernel_launch(void* const* d_in, const int* in_sizes, int n_in,
                              void* d_out, int out_size, void* d_ws,
                              size_t ws_size, hipStream_t stream) {
  (void)in_sizes; (void)n_in; (void)out_size; (void)ws_size;
  const float* input = (const float*)d_in[0];  // [512, 64, 32]
  const float* W0 = (const float*)d_in[1];     // [128, 4096]
  const float* b0 = (const float*)d_in[2];     // [128]
  const float* W1 = (const float*)d_in[3];     // [128, 8192]
  const float* b1 = (const float*)d_in[4];     // [128]
  const float* W2 = (const float*)d_in[5];     // [128, 8192]
  const float* b2 = (const float*)d_in[6];     // [128]
  float* out = (float*)d_out;                  // [512, 384]

  constexpr int B = 512;
  float* x1 = (float*)d_ws;                    // [512, 128, 32] = 8 MB
  float* x2 = x1 + (size_t)B * 128 * 32;       // [512, 128, 32] = 8 MB

  dim3 grid(B / 2), block(256);
  fused_bilinear_layer<64, true><<<grid, block, 0, stream>>>(
      input, input, W0, b0, x1, out, 0);
  fused_bilinear_layer<128, true><<<grid, block, 0, stream>>>(
      x1, input, W1, b1, x2, out, 128);
  fused_bilinear_layer<128, false><<<grid, block, 0, stream>>>(
      x2, input, W2, b2, nullptr, out, 256);
}